// QuantumAttention_47880295416061
// MI455X (gfx1250) — compile-verified
//
#include <hip/hip_runtime.h>
#include <math.h>

typedef float v2f __attribute__((ext_vector_type(2)));
typedef float v8f __attribute__((ext_vector_type(8)));

#define NQ 10
#define SDIM 1024           // 2^NQ
#define NTOK 2048           // B*L = 4*512
#define ROWF (2*SDIM)       // interleaved (re,im) row length = 2048 floats

__device__ __forceinline__ float2 cmul(float2 a, float2 b) {
    return make_float2(a.x*b.x - a.y*b.y, a.x*b.y + a.y*b.x);
}
__device__ __forceinline__ float2 cadd(float2 a, float2 b) {
    return make_float2(a.x + b.x, a.y + b.y);
}

// ---------------------------------------------------------------------------
// Phase 1: statevector preparation, one workgroup (128 thr) per token.
// Circuit closed form:
//   psi = D2(W) . CX . (tensor U(W)) . D1(data+pos) . CX . (tensor v_q)
// with v_q = U(pos_q) U(data_q) H|0>, CX chain = Gray-code permutation
// (post-CX index y reads pre-CX index x = y ^ (y>>1)), D* diagonal phases.
// Only the U(W) layer needs butterfly sweeps over the LDS state.
// Output: interleaved rows  ws[token][2*s+0]=re, [2*s+1]=im  (len 2048).
// ---------------------------------------------------------------------------
__global__ __launch_bounds__(128)
void qa_prepare(const float* __restrict__ ydata, const float* __restrict__ ypos,
                const float* __restrict__ xdata, const float* __restrict__ xpos,
                const float* __restrict__ Wq,   const float* __restrict__ Wk,
                float* __restrict__ Qws, float* __restrict__ Kws)
{
    __shared__ float2 st[SDIM];          // statevector (8 KB)
    __shared__ float2 vq[NQ][2];         // per-qubit product vectors
    __shared__ float2 gW[NQ][4];         // U(W_q) 2x2 complex gates
    __shared__ float  th1[NQ];           // data+pos (fused RZ angles)
    __shared__ float  wv[NQ];            // W angles

    const int t   = blockIdx.x;          // 0..4095
    const bool isQ = (t < NTOK);
    const int tok = isQ ? t : (t - NTOK);
    const float* data = (isQ ? ydata : xdata) + tok * NQ;
    const float* pos  = (isQ ? ypos  : xpos ) + tok * NQ;
    const float* W    = isQ ? Wq : Wk;
    float* outp = (isQ ? Qws : Kws) + (size_t)tok * ROWF;

    const int tid = threadIdx.x;

    if (tid < NQ) {
        const int q = tid;
        const float d = data[q], p = pos[q], w = W[q];
        float sd, cd; sincosf(0.5f * d, &sd, &cd);
        float sp, cp; sincosf(0.5f * p, &sp, &cp);
        float sw, cw; sincosf(0.5f * w, &sw, &cw);
        // u(theta): g00=c*e^{-it/2} g01=-s*e^{+it/2} g10=s*e^{-it/2} g11=c*e^{+it/2}
        float2 gd[4] = { make_float2(cd*cd, -cd*sd), make_float2(-sd*cd, -sd*sd),
                         make_float2(sd*cd, -sd*sd), make_float2( cd*cd,  cd*sd) };
        float2 gp[4] = { make_float2(cp*cp, -cp*sp), make_float2(-sp*cp, -sp*sp),
                         make_float2(sp*cp, -sp*sp), make_float2( cp*cp,  cp*sp) };
        // a = u(d) * (r, r),  r = 1/sqrt(2)
        const float r = 0.70710678118654752440f;
        float2 a0 = make_float2(r*(gd[0].x + gd[1].x), r*(gd[0].y + gd[1].y));
        float2 a1 = make_float2(r*(gd[2].x + gd[3].x), r*(gd[2].y + gd[3].y));
        vq[q][0] = cadd(cmul(gp[0], a0), cmul(gp[1], a1));
        vq[q][1] = cadd(cmul(gp[2], a0), cmul(gp[3], a1));
        gW[q][0] = make_float2(cw*cw, -cw*sw);
        gW[q][1] = make_float2(-sw*cw, -sw*sw);
        gW[q][2] = make_float2(sw*cw, -sw*sw);
        gW[q][3] = make_float2(cw*cw,  cw*sw);
        th1[q] = d + p;
        wv[q]  = w;
    }
    __syncthreads();

    float p1b = 0.f, p2b = 0.f;
#pragma unroll
    for (int q = 0; q < NQ; ++q) { p1b -= 0.5f * th1[q]; p2b -= 0.5f * wv[q]; }

    // init: state after first CX chain + fused RZ(data+pos) diagonal
#pragma unroll
    for (int j = 0; j < 8; ++j) {
        const int y = tid + j * 128;
        const int x = y ^ (y >> 1);          // pre-CX index (Gray decode)
        float2 acc = make_float2(1.f, 0.f);
        float phi = p1b;
#pragma unroll
        for (int q = 0; q < NQ; ++q) {
            const int pbit = NQ - 1 - q;     // qubit 0 is MSB
            acc = cmul(acc, vq[q][(x >> pbit) & 1]);
            if ((y >> pbit) & 1) phi += th1[q];
        }
        float sph, cph; sincosf(phi, &sph, &cph);
        st[y] = cmul(acc, make_float2(cph, sph));
    }

    // apply U(W_q) butterflies
    for (int q = 0; q < NQ; ++q) {
        __syncthreads();
        const int pbit = NQ - 1 - q;
        const float2 g0 = gW[q][0], g1 = gW[q][1], g2 = gW[q][2], g3 = gW[q][3];
#pragma unroll
        for (int j = 0; j < 4; ++j) {
            const int i  = tid + j * 128;    // 512 pairs
            const int i0 = ((i >> pbit) << (pbit + 1)) | (i & ((1 << pbit) - 1));
            const int i1 = i0 | (1 << pbit);
            const float2 a0 = st[i0], a1 = st[i1];
            st[i0] = cadd(cmul(g0, a0), cmul(g1, a1));
            st[i1] = cadd(cmul(g2, a0), cmul(g3, a1));
        }
    }
    __syncthreads();

    // final CX chain (index permutation) + RZ(W) phase, write interleaved
#pragma unroll
    for (int j = 0; j < 8; ++j) {
        const int z = tid + j * 128;
        const int x = z ^ (z >> 1);
        float phi = p2b;
#pragma unroll
        for (int q = 0; q < NQ; ++q)
            if ((z >> (NQ - 1 - q)) & 1) phi += wv[q];
        float sph, cph; sincosf(phi, &sph, &cph);
        const float2 psi = cmul(st[x], make_float2(cph, sph));
        ((float2*)outp)[z] = psi;
    }
}

// ---------------------------------------------------------------------------
// Phase 2: per-batch complex GEMM  amp = Q K^H  via V_WMMA_F32_16X16X4_F32,
// out = |amp|^2.  Interleaved rows (len 2048):
//   amp_r = <[qr,qi],[kr,ki]>      -> A1 = Q row
//   amp_i = <[qi,-qr],[kr,ki]>     -> A2 = per-pair swizzle of A1
// A-frag lane holds an (even,odd) k pair -> A2 = (A1.y, -A1.x) in registers.
// Block tile 64x64, 4 waves each compute 32x32 as 2x2 WMMA subtiles, KC=64.
// ---------------------------------------------------------------------------
#define BM 64
#define BN 64
#define KC 64
#define LDT 68   // padded LDS stride (mult of 4 floats -> 16B-aligned rows)

__global__ __launch_bounds__(128)
void qa_gemm(const float* __restrict__ Qws, const float* __restrict__ Kws,
             float* __restrict__ out)
{
    __shared__ float sQ[BM][LDT];
    __shared__ float sK[BN][LDT];

    const int bid = blockIdx.x;          // 256 blocks: b(4) x mt(8) x nt(8)
    const int b  = bid >> 6;
    const int mt = (bid >> 3) & 7;
    const int nt = bid & 7;
    const int mBase = mt * BM, nBase = nt * BN;

    const float* Qb = Qws + ((size_t)b * 512 + mBase) * ROWF;
    const float* Kb = Kws + ((size_t)b * 512 + nBase) * ROWF;

    const int tid  = threadIdx.x;
    const int wave = tid >> 5;
    const int lane = tid & 31;
    const int half = lane >> 4;          // K sub-pair select
    const int lr   = lane & 15;          // M (A) / N (B) row within 16
    const int wm   = wave >> 1, wn = wave & 1;

    v8f accR[2][2], accI[2][2];
#pragma unroll
    for (int mi = 0; mi < 2; ++mi)
#pragma unroll
        for (int ni = 0; ni < 2; ++ni) { accR[mi][ni] = (v8f)0.f; accI[mi][ni] = (v8f)0.f; }

    const int lrow = tid >> 4;           // 0..7
    const int lcol = (tid & 15) << 2;    // 0..60, float4 columns

    for (int kc = 0; kc < ROWF; kc += KC) {
        __syncthreads();
#pragma unroll
        for (int rr = 0; rr < 8; ++rr) {
            const int row = lrow + rr * 8;
            const float4 q4 = *(const float4*)(Qb + (size_t)row * ROWF + kc + lcol);
            const float4 k4 = *(const float4*)(Kb + (size_t)row * ROWF + kc + lcol);
            *(float4*)&sQ[row][lcol] = q4;
            *(float4*)&sK[row][lcol] = k4;
        }
        __syncthreads();

#pragma unroll
        for (int kk = 0; kk < KC; kk += 4) {
            v2f A1[2], Bf[2];
#pragma unroll
            for (int mi = 0; mi < 2; ++mi) {
                const float2 v = *(const float2*)&sQ[wm * 32 + mi * 16 + lr][kk + half * 2];
                A1[mi].x = v.x; A1[mi].y = v.y;
            }
#pragma unroll
            for (int ni = 0; ni < 2; ++ni) {
                const float2 v = *(const float2*)&sK[wn * 32 + ni * 16 + lr][kk + half * 2];
                Bf[ni].x = v.x; Bf[ni].y = v.y;
            }
#pragma unroll
            for (int mi = 0; mi < 2; ++mi) {
                v2f A2; A2.x = A1[mi].y; A2.y = -A1[mi].x;   // [qi,-qr] pair swizzle
#pragma unroll
                for (int ni = 0; ni < 2; ++ni) {
                    accR[mi][ni] = __builtin_amdgcn_wmma_f32_16x16x4_f32(
                        false, A1[mi], false, Bf[ni], (short)0, accR[mi][ni], false, false);
                    accI[mi][ni] = __builtin_amdgcn_wmma_f32_16x16x4_f32(
                        false, A2,     false, Bf[ni], (short)0, accI[mi][ni], false, false);
                }
            }
        }
    }

    // epilogue: |amp|^2 ; C/D layout: vgpr e -> M = e + 8*half, lane lr -> N
    const size_t outBase = (size_t)b * 512 * 512;
#pragma unroll
    for (int mi = 0; mi < 2; ++mi)
#pragma unroll
        for (int ni = 0; ni < 2; ++ni) {
            const int n = nBase + wn * 32 + ni * 16 + lr;
#pragma unroll
            for (int e = 0; e < 8; ++e) {
                const int m = mBase + wm * 32 + mi * 16 + half * 8 + e;
                const float rr = accR[mi][ni][e];
                const float ii = accI[mi][ni][e];
                out[outBase + (size_t)m * 512 + n] = rr * rr + ii * ii;
            }
        }
}

// ---------------------------------------------------------------------------
extern "C" void kernel_launch(void* const* d_in, const int* in_sizes, int n_in,
                              void* d_out, int out_size, void* d_ws, size_t ws_size,
                              hipStream_t stream)
{
    const float* y_data = (const float*)d_in[0];
    const float* y_pos  = (const float*)d_in[1];
    const float* x_data = (const float*)d_in[2];
    const float* x_pos  = (const float*)d_in[3];
    const float* W_q    = (const float*)d_in[4];
    const float* W_k    = (const float*)d_in[5];

    float* Qws = (float*)d_ws;                       // 2048 tokens x 2048 floats = 16 MB
    float* Kws = Qws + (size_t)NTOK * ROWF;          // + 16 MB

    qa_prepare<<<2 * NTOK, 128, 0, stream>>>(y_data, y_pos, x_data, x_pos,
                                             W_q, W_k, Qws, Kws);
    qa_gemm<<<256, 128, 0, stream>>>(Qws, Kws, (float*)d_out);
}